// Net_48773648614109
// MI455X (gfx1250) — compile-verified
//
#include <hip/hip_runtime.h>

typedef __attribute__((ext_vector_type(2))) float v2f;
typedef __attribute__((ext_vector_type(8))) float v8f;

#define EMB_DIM          64
#define PAIRS_PER_WAVE   16
#define WAVES_PER_BLOCK  8
#define BLOCK_THREADS    (WAVES_PER_BLOCK * 32)

// One wave computes 16 dot products (dim 64) via 16x V_WMMA_F32_16X16X4_F32:
// diag(U * V^T) lands on the accumulator diagonal, then -logsigmoid + reduce.
__global__ __launch_bounds__(BLOCK_THREADS)
void sgns_loss_wmma(const int* __restrict__ pos_u, const int* __restrict__ pos_v,
                    const int* __restrict__ neg_u, const int* __restrict__ neg_v,
                    const float* __restrict__ u_emb, const float* __restrict__ v_emb,
                    float* __restrict__ out, int B, int total)
{
    const int lane = threadIdx.x & 31;
    const int wave = threadIdx.x >> 5;
    const int waveBase = (blockIdx.x * WAVES_PER_BLOCK + wave) * PAIRS_PER_WAVE;
    if (waveBase >= total) return;              // wave-uniform exit

    const int r = lane & 15;                    // pair row served by this lane
    const int h = lane >> 4;                    // half selects K-offset +2
    int p = waveBase + r;
    const bool valid = (p < total);
    p = valid ? p : (total - 1);                // clamp, mask contribution later

    // Pair -> (u index, v index). p < B: positive edge; else negative edge.
    int uidx, vidx;
    if (p < B) { uidx = pos_u[p];     vidx = pos_v[p];     }
    else       { uidx = neg_u[p - B]; vidx = neg_v[p - B]; }

    const float* __restrict__ urow = u_emb + (size_t)uidx * EMB_DIM + 2 * h;
    const float* __restrict__ vrow = v_emb + (size_t)vidx * EMB_DIM + 2 * h;

    // Gather A (u rows) and B (v rows) chunks: 16 float2 loads each,
    // 8B aligned (rows are 256B aligned), -> global_load_b64 gathers.
    v2f a[16], b[16];
#pragma unroll
    for (int t = 0; t < 16; ++t) {
        a[t] = *(const v2f*)(urow + 4 * t);
        b[t] = *(const v2f*)(vrow + 4 * t);
    }

    // 16 WMMA steps accumulate the full K=64 dot products on the diagonal.
    v8f c = {};
#pragma unroll
    for (int t = 0; t < 16; ++t) {
        c = __builtin_amdgcn_wmma_f32_16x16x4_f32(
                /*neg_a=*/false, a[t], /*neg_b=*/false, b[t],
                /*c_mod=*/(short)0, c, /*reuse_a=*/false, /*reuse_b=*/false);
    }

    // Diagonal extraction per ISA C/D layout:
    //   m=0..7  -> lane m,    VGPR m
    //   m=8..15 -> lane m+16, VGPR m-8
    const int j = (lane < 16) ? lane : (lane - 24);   // 0..7 on active lanes
    float d = c[0];
#pragma unroll
    for (int i = 1; i < 8; ++i) d = (j == i) ? c[i] : d;

    float contrib = 0.0f;
    const bool active = (lane < 8) || (lane >= 24);
    if (active && valid) {
        // -logsigmoid(x) = log1p(exp(-|x|)) - min(x, 0)   (numerically stable)
        const float x = d;
        contrib = log1pf(__expf(-fabsf(x))) - fminf(x, 0.0f);
    }

    // wave32 reduction, then one f32 atomic per wave.
    float s = contrib;
#pragma unroll
    for (int off = 16; off > 0; off >>= 1)
        s += __shfl_xor(s, off, 32);
    if (lane == 0)
        atomicAdd(out, s);
}

extern "C" void kernel_launch(void* const* d_in, const int* in_sizes, int n_in,
                              void* d_out, int out_size, void* d_ws, size_t ws_size,
                              hipStream_t stream) {
    const int*   pos_u = (const int*)d_in[0];
    const int*   pos_v = (const int*)d_in[1];
    const int*   neg_u = (const int*)d_in[2];
    const int*   neg_v = (const int*)d_in[3];
    const float* u_emb = (const float*)d_in[4];
    const float* v_emb = (const float*)d_in[5];
    float* out = (float*)d_out;

    const int B     = in_sizes[0];       // 16384
    const int BK    = in_sizes[2];       // 16384 * 20
    const int total = B + BK;            // 344064 (multiple of 16)

    hipMemsetAsync(out, 0, sizeof(float), stream);   // graph-capture safe

    const int pairsPerBlock = PAIRS_PER_WAVE * WAVES_PER_BLOCK;   // 128
    const int grid = (total + pairsPerBlock - 1) / pairsPerBlock; // 2688
    sgns_loss_wmma<<<grid, BLOCK_THREADS, 0, stream>>>(
        pos_u, pos_v, neg_u, neg_v, u_emb, v_emb, out, B, total);
}